// GeneralGNN_28750511079706
// MI455X (gfx1250) — compile-verified
//
#include <hip/hip_runtime.h>

typedef __attribute__((ext_vector_type(2))) float v2f;
typedef __attribute__((ext_vector_type(8))) float v8f;

#define DD 256           // feature dim (compile-time for tiling)
#define LDS_STRIDE 260   // 256 + 4 pad floats -> conflict-free A-fragment reads

// ---------------- utility kernels ----------------

__global__ void zero_kernel(float* p, long n) {
  long t = (long)blockIdx.x * blockDim.x + threadIdx.x;
  if (t < n) p[t] = 0.0f;
}

__global__ void transpose2_kernel(const float* __restrict__ W1, const float* __restrict__ W2,
                                  float* __restrict__ Wt1, float* __restrict__ Wt2) {
  int t = blockIdx.x * blockDim.x + threadIdx.x;   // DD*DD threads
  int r = t >> 8, c = t & (DD - 1);
  Wt1[c * DD + r] = W1[r * DD + c];
  Wt2[c * DD + r] = W2[r * DD + c];
}

// feats[n,c] = x[b, left[n], c] + x[b, right[n], c]
__global__ void feats_kernel(const float* __restrict__ x, const int* __restrict__ nb,
                             const int* __restrict__ nl, const int* __restrict__ nr,
                             float* __restrict__ feats, int Nn, int L) {
  long t = (long)blockIdx.x * blockDim.x + threadIdx.x;
  long total = (long)Nn * DD;
  if (t >= total) return;
  int n = (int)(t >> 8), c = (int)(t & (DD - 1));
  long base = (long)nb[n] * L;
  feats[t] = x[(base + nl[n]) * DD + c] + x[(base + nr[n]) * DD + c];
}

__global__ void deg_kernel(const int* __restrict__ dst, int* __restrict__ degcnt, int E) {
  int t = blockIdx.x * blockDim.x + threadIdx.x;
  if (t < E) atomicAdd(&degcnt[dst[t]], 1);
}

// agg[dst] += h[src], one 256-wide row per block -> coalesced, native f32 atomics
__global__ void scatter_kernel(const float* __restrict__ h, const int* __restrict__ src,
                               const int* __restrict__ dst, float* __restrict__ agg, int E) {
  long t = (long)blockIdx.x * blockDim.x + threadIdx.x;
  if (t >= (long)E * DD) return;
  int e = (int)(t >> 8), c = (int)(t & (DD - 1));
  unsafeAtomicAdd(&agg[(long)dst[e] * DD + c], h[(long)src[e] * DD + c]);
}

// ---------------- fused normalize + GEMM (fp32 WMMA) ----------------
// out[r0..r0+31, :] = act( ((h + scat)/deg)[r0..r0+31, :] @ W )  with Wt = W^T
template <bool RELU>
__global__ __launch_bounds__(256) void gemm_gcn(const float* __restrict__ h,
                                                const float* __restrict__ scat,
                                                const int* __restrict__ degcnt,
                                                const float* __restrict__ Wt,
                                                float* __restrict__ outp, int Nn) {
  __shared__ float lds[32 * LDS_STRIDE];
  const int r0 = blockIdx.x * 32;
  const int tid = threadIdx.x;

  // Stage normalized input rows [32 x 256] into LDS (coalesced, clamped rows).
  for (int i = 0; i < 32; ++i) {
    int row = r0 + i;
    int rc = row < Nn ? row : Nn - 1;
    float inv = 1.0f / (float)(degcnt[rc] + 1);
    lds[i * LDS_STRIDE + tid] =
        (h[(long)rc * DD + tid] + scat[(long)rc * DD + tid]) * inv;
  }
  __syncthreads();

  const int lane = tid & 31;
  const int w = tid >> 5;        // 8 waves
  const int rtile = w & 1;       // 2 row tiles of 16
  const int cgrp = w >> 1;       // 4 col groups of 64
  const int hi = lane >> 4;      // lane half selects K pair
  const int m = lane & 15;       // A: row within tile; B: col within tile

  v8f acc0 = {}, acc1 = {}, acc2 = {}, acc3 = {};
  const float* wt0 = Wt + (long)(cgrp * 64 + 0 * 16 + m) * DD;
  const float* wt1 = Wt + (long)(cgrp * 64 + 1 * 16 + m) * DD;
  const float* wt2 = Wt + (long)(cgrp * 64 + 2 * 16 + m) * DD;
  const float* wt3 = Wt + (long)(cgrp * 64 + 3 * 16 + m) * DD;
  const float* arow = &lds[(rtile * 16 + m) * LDS_STRIDE];

#pragma unroll 4
  for (int ks = 0; ks < DD / 4; ++ks) {
    int kk = ks * 4 + 2 * hi;                       // lane-half K offset
    v2f a = *(const v2f*)(arow + kk);               // A[m, kk..kk+1] (ds b64)
    v2f b0 = *(const v2f*)(wt0 + kk);               // B[kk..kk+1, n] via Wt
    v2f b1 = *(const v2f*)(wt1 + kk);
    v2f b2 = *(const v2f*)(wt2 + kk);
    v2f b3 = *(const v2f*)(wt3 + kk);
    acc0 = __builtin_amdgcn_wmma_f32_16x16x4_f32(false, a, false, b0, (short)0, acc0, false, false);
    acc1 = __builtin_amdgcn_wmma_f32_16x16x4_f32(false, a, false, b1, (short)0, acc1, false, false);
    acc2 = __builtin_amdgcn_wmma_f32_16x16x4_f32(false, a, false, b2, (short)0, acc2, false, false);
    acc3 = __builtin_amdgcn_wmma_f32_16x16x4_f32(false, a, false, b3, (short)0, acc3, false, false);
  }

  // C/D layout: VGPR v -> M = v + 8*hi, N = lane&15
  const int colb = cgrp * 64 + m;
#pragma unroll
  for (int v = 0; v < 8; ++v) {
    int row = r0 + rtile * 16 + v + 8 * hi;
    if (row < Nn) {
      float o0 = acc0[v], o1 = acc1[v], o2 = acc2[v], o3 = acc3[v];
      if (RELU) {
        o0 = fmaxf(o0, 0.0f); o1 = fmaxf(o1, 0.0f);
        o2 = fmaxf(o2, 0.0f); o3 = fmaxf(o3, 0.0f);
      }
      float* orow = outp + (long)row * DD + colb;
      orow[0]  = o0;
      orow[16] = o1;
      orow[32] = o2;
      orow[48] = o3;
    }
  }
}

// ---------------- global mean pool + broadcast add ----------------

__global__ void pool_kernel(const float* __restrict__ outp, float* __restrict__ gsum, int M) {
  int b = blockIdx.x;
  int c = threadIdx.x;
  const float* p = outp + (long)b * M * DD + c;
  float s = 0.0f;
  for (int mm = 0; mm < M; ++mm) s += p[(long)mm * DD];
  gsum[b * DD + c] = s / (float)M;
}

__global__ void add_kernel(float* __restrict__ outp, const float* __restrict__ gsum,
                           int Nn, int M) {
  long t = (long)blockIdx.x * blockDim.x + threadIdx.x;
  if (t >= (long)Nn * DD) return;
  int n = (int)(t >> 8);
  int b = n / M;
  outp[t] += gsum[b * DD + (t & (DD - 1))];
}

// ---------------- launcher ----------------

extern "C" void kernel_launch(void* const* d_in, const int* in_sizes, int n_in,
                              void* d_out, int out_size, void* d_ws, size_t ws_size,
                              hipStream_t stream) {
  const float* x        = (const float*)d_in[0];
  const int* node_batch = (const int*)d_in[1];
  const int* node_left  = (const int*)d_in[2];
  const int* node_right = (const int*)d_in[3];
  const int* edge_src   = (const int*)d_in[4];
  const int* edge_dst   = (const int*)d_in[5];
  const float* W1       = (const float*)d_in[6];
  const float* W2       = (const float*)d_in[7];

  const int Nn = in_sizes[1];               // total nodes B*M
  const int E  = in_sizes[4];               // total edges
  const long BL = (long)in_sizes[0] / DD;   // B*L
  const int Bb = (int)(2 * BL - (long)Nn);  // from N = B*(2L-1)
  const int L  = (int)(BL / Bb);
  const int M  = Nn / Bb;

  // workspace carve-up
  float* bufA   = (float*)d_ws;                      // [Nn, DD] feats -> h1
  float* bufB   = bufA + (long)Nn * DD;              // [Nn, DD] scatter accumulator
  int*   degcnt = (int*)(bufB + (long)Nn * DD);      // [Nn]
  float* Wt1    = (float*)(degcnt + Nn);             // [DD, DD]
  float* Wt2    = Wt1 + DD * DD;                     // [DD, DD]
  float* gsum   = Wt2 + DD * DD;                     // [Bb, DD]

  // 1) transpose weights (B-fragments become contiguous b64 loads)
  transpose2_kernel<<<DD * DD / 256, 256, 0, stream>>>(W1, W2, Wt1, Wt2);

  // 2) node features
  feats_kernel<<<Nn, 256, 0, stream>>>(x, node_batch, node_left, node_right, bufA, Nn, L);

  // 3) zero scatter buffer + degree counters (contiguous 4-byte words)
  {
    long zc = (long)Nn * DD + Nn;
    zero_kernel<<<(unsigned)((zc + 255) / 256), 256, 0, stream>>>(bufB, zc);
  }

  // 4) degrees
  deg_kernel<<<(E + 255) / 256, 256, 0, stream>>>(edge_dst, degcnt, E);

  // 5) layer-1 scatter then fused normalize+GEMM+ReLU (in place into bufA)
  scatter_kernel<<<E, 256, 0, stream>>>(bufA, edge_src, edge_dst, bufB, E);
  gemm_gcn<true><<<(Nn + 31) / 32, 256, 0, stream>>>(bufA, bufB, degcnt, Wt1, bufA, Nn);

  // 6) layer-2 scatter then GEMM straight into d_out
  {
    long zc = (long)Nn * DD;
    zero_kernel<<<(unsigned)((zc + 255) / 256), 256, 0, stream>>>(bufB, zc);
  }
  scatter_kernel<<<E, 256, 0, stream>>>(bufA, edge_src, edge_dst, bufB, E);
  gemm_gcn<false><<<(Nn + 31) / 32, 256, 0, stream>>>(bufA, bufB, degcnt, Wt2,
                                                      (float*)d_out, Nn);

  // 7) per-graph mean pool + broadcast add
  pool_kernel<<<Bb, 256, 0, stream>>>((const float*)d_out, gsum, M);
  add_kernel<<<(unsigned)(((long)Nn * DD + 255) / 256), 256, 0, stream>>>(
      (float*)d_out, gsum, Nn, M);
}